// CQLoss_84756884619454
// MI455X (gfx1250) — compile-verified
//
#include <hip/hip_runtime.h>

// ---------------- problem constants (from reference) ----------------
static constexpr int B = 64, S = 128, D = 2048, P = 118, C = 2, V = 512;
static constexpr int PC   = P * C;          // 236 floats per (b,s) row
static constexpr int ROWV = PC / 4;         // 59 float4 per row
static constexpr int DV   = D / 4;          // 512 float4 per row

static constexpr long long N_Z  = (long long)B * S * D;      // 16,777,216
static constexpr long long N_PT = (long long)B * S * P * C;  //  1,933,312
static constexpr long long N_QY = (long long)B * S * V;      //  4,194,304

static constexpr int N4_Z    = (int)(N_Z  / 4);  // 4,194,304
static constexpr int N4_PT   = (int)(N_PT / 4);  //   483,328
static constexpr int N4_QY   = (int)(N_QY / 4);  // 1,048,576
static constexpr int N4_BEST = B * PC / 4;       //     3,776

static constexpr int BLK  = 256;   // 8 wave32s
static constexpr int ITER = 8;     // float4s per thread

static constexpr int NB_KLD = N4_QY / (BLK * ITER);  // 512
static constexpr int NB_REC = N4_Z  / (BLK * ITER);  // 2048
static constexpr int NB_PTS = N4_PT / (BLK * ITER);  // 236

static_assert(NB_KLD * BLK * ITER == N4_QY, "kld grid exact");
static_assert(NB_REC * BLK * ITER == N4_Z,  "recon grid exact");
static_assert(NB_PTS * BLK * ITER == N4_PT, "pts grid exact");

// ws layout (floats): [kld NB_KLD][recon NB_REC][disk NB_PTS][land NB_PTS][best 1][bestland 1]
static constexpr int WS_KLD  = 0;
static constexpr int WS_REC  = WS_KLD + NB_KLD;
static constexpr int WS_DISK = WS_REC + NB_REC;
static constexpr int WS_LAND = WS_DISK + NB_PTS;
static constexpr int WS_BEST = WS_LAND + NB_PTS;   // 2 floats

// ---------------- WMMA wave reduction (CDNA5 f32 16x16x4) ----------------
typedef float v2f __attribute__((ext_vector_type(2)));
typedef float v8f __attribute__((ext_vector_type(8)));

// Sum 32 lane values with one V_WMMA_F32_16X16X4_F32:
// A[m,k] holds lane partials (lane L<16 -> A[L][0], lane L>=16 -> A[L-16][2]),
// B = all-ones, so D rows are A row-sums; summing 8 C/D VGPRs gives the
// half-wave totals, one xor-16 shuffle finishes.  EXEC must be all ones.
__device__ __forceinline__ float wave_sum_wmma(float v) {
    v2f a;    a[0] = v;    a[1] = 0.0f;
    v2f ones; ones[0] = 1.0f; ones[1] = 1.0f;
    v8f c = {};
    c = __builtin_amdgcn_wmma_f32_16x16x4_f32(
            /*neg_a=*/false, a, /*neg_b=*/false, ones,
            /*c_mod=*/(short)0, c, /*reuse_a=*/false, /*reuse_b=*/false);
    float s = c[0] + c[1] + c[2] + c[3] + c[4] + c[5] + c[6] + c[7];
    s += __shfl_xor(s, 16, 32);   // combine the two half-wave sums
    return s;
}

// Full 256-thread block reduction; result written by thread 0 to *slot.
// Callable multiple times per kernel (leading barrier protects smem reuse).
__device__ __forceinline__ void block_reduce_to(float v, float* __restrict__ slot) {
    __shared__ float smem[BLK / 32];
    __syncthreads();
    float s = wave_sum_wmma(v);
    const int lane = threadIdx.x & 31;
    const int wave = threadIdx.x >> 5;
    if (lane == 0) smem[wave] = s;
    __syncthreads();
    if (threadIdx.x == 0) {
        float t = 0.0f;
#pragma unroll
        for (int i = 0; i < BLK / 32; ++i) t += smem[i];
        *slot = t;
    }
}

// ---------------- reduction kernels ----------------
__global__ void __launch_bounds__(BLK)
kld_kernel(const float4* __restrict__ qy, float* __restrict__ out) {
    const float log_g = -6.2383246250395075f;   // ln(1/512)
    int idx = blockIdx.x * (BLK * ITER) + threadIdx.x;
    float acc = 0.0f;
#pragma unroll
    for (int i = 0; i < ITER; ++i, idx += BLK) {
        float4 q = qy[idx];
        acc += q.x * (logf(q.x + 1e-20f) - log_g);
        acc += q.y * (logf(q.y + 1e-20f) - log_g);
        acc += q.z * (logf(q.z + 1e-20f) - log_g);
        acc += q.w * (logf(q.w + 1e-20f) - log_g);
    }
    block_reduce_to(acc, &out[blockIdx.x]);
}

__global__ void __launch_bounds__(BLK)
recon_kernel(const float4* __restrict__ zs, const float4* __restrict__ rzs,
             const long long* __restrict__ mapping, float* __restrict__ out) {
    int idx = blockIdx.x * (BLK * ITER) + threadIdx.x;
    float acc = 0.0f;
#pragma unroll
    for (int i = 0; i < ITER; ++i, idx += BLK) {
        const int row    = idx >> 9;          // / DV (512 float4 per row)
        const int within = idx & (DV - 1);
        const int b      = row >> 7;          // / S (128)
        const int m      = (int)mapping[row];
        const float4 z = zs[idx];
        const float4 r = rzs[(size_t)(b * S + m) * DV + within];
        const float dx = r.x - z.x, dy = r.y - z.y;
        const float dz = r.z - z.z, dw = r.w - z.w;
        acc += dx * dx + dy * dy + dz * dz + dw * dw;
    }
    block_reduce_to(acc, &out[blockIdx.x]);
}

__global__ void __launch_bounds__(BLK)
pts_kernel(const float4* __restrict__ pts, const float4* __restrict__ pts_gt,
           const long long* __restrict__ mapping,
           float* __restrict__ out_disk, float* __restrict__ out_land) {
    int idx = blockIdx.x * (BLK * ITER) + threadIdx.x;
    float dacc = 0.0f, lacc = 0.0f;
#pragma unroll
    for (int i = 0; i < ITER; ++i, idx += BLK) {
        const int row    = idx / ROWV;        // magic-mul division by 59
        const int within = idx - row * ROWV;
        const int b      = row >> 7;
        const int m      = (int)mapping[row];
        const float4 x = pts[(size_t)(b * S + m) * ROWV + within];
        const float4 g = pts_gt[idx];
        const float dx = x.x - g.x, dy = x.y - g.y;
        const float dz = x.z - g.z, dw = x.w - g.w;
        const float s01 = dx * dx + dy * dy;
        const float s23 = dz * dz + dw * dw;
        dacc += s01 + s23;
        // landmark p in {0,29,88,117}: vec4 'within' covers p=2w (lo pair), 2w+1 (hi pair)
        if (within == 0  || within == 44) lacc += s01;   // p = 0, 88
        if (within == 14 || within == 58) lacc += s23;   // p = 29, 117
    }
    block_reduce_to(dacc, &out_disk[blockIdx.x]);
    block_reduce_to(lacc, &out_land[blockIdx.x]);
}

__global__ void __launch_bounds__(BLK)
best_kernel(const float4* __restrict__ best, const float4* __restrict__ best_gt,
            float* __restrict__ out2) {
    float facc = 0.0f, lacc = 0.0f;
    for (int idx = threadIdx.x; idx < N4_BEST; idx += BLK) {
        const int row    = idx / ROWV;
        const int within = idx - row * ROWV;
        const float4 x = best[idx];
        const float4 g = best_gt[idx];
        const float dx = x.x - g.x, dy = x.y - g.y;
        const float dz = x.z - g.z, dw = x.w - g.w;
        const float s01 = dx * dx + dy * dy;
        const float s23 = dz * dz + dw * dw;
        facc += s01 + s23;
        if (within == 0  || within == 44) lacc += s01;
        if (within == 14 || within == 58) lacc += s23;
    }
    block_reduce_to(facc, &out2[0]);
    block_reduce_to(lacc, &out2[1]);
}

// ---------------- deterministic finalize (f64 combine) ----------------
__global__ void __launch_bounds__(BLK)
finalize_kernel(const float* __restrict__ ws, float* __restrict__ out) {
    __shared__ double sm[BLK];
    auto rsum = [&](const float* p, int n) -> double {
        double s = 0.0;
        for (int i = threadIdx.x; i < n; i += BLK) s += (double)p[i];
        sm[threadIdx.x] = s;
        __syncthreads();
        for (int off = BLK / 2; off > 0; off >>= 1) {
            if ((int)threadIdx.x < off) sm[threadIdx.x] += sm[threadIdx.x + off];
            __syncthreads();
        }
        double r = sm[0];
        __syncthreads();
        return r;
    };
    const double skld  = rsum(ws + WS_KLD,  NB_KLD);
    const double srec  = rsum(ws + WS_REC,  NB_REC);
    const double sdisk = rsum(ws + WS_DISK, NB_PTS);
    const double sland = rsum(ws + WS_LAND, NB_PTS);
    if (threadIdx.x == 0) {
        const double sbest  = (double)ws[WS_BEST + 0];
        const double sbland = (double)ws[WS_BEST + 1];
        const double kld   = skld  / (double)(B * S);
        const double recon = srec  / (double)N_Z;
        const double disk  = sdisk / (double)N_PT;
        const double land  = sland / (double)((long long)B * S * 4 * C);
        const double bm    = sbest  / (double)(B * P * C);
        const double bl    = sbland / (double)(B * 4 * C);
        out[0] = (float)(0.1 * kld + 1.0 * recon
                         + disk + 10.0 * land
                         + bm   + 10.0 * bl);
    }
}

// ---------------- launch ----------------
extern "C" void kernel_launch(void* const* d_in, const int* in_sizes, int n_in,
                              void* d_out, int out_size, void* d_ws, size_t ws_size,
                              hipStream_t stream) {
    (void)in_sizes; (void)n_in; (void)out_size; (void)ws_size;
    const float4*    zs      = (const float4*)d_in[0];
    const float4*    rzs     = (const float4*)d_in[1];
    const float4*    pts     = (const float4*)d_in[2];
    const float4*    pts_gt  = (const float4*)d_in[3];
    const float4*    qy      = (const float4*)d_in[4];
    // d_in[5] = logits (unused by reference)
    const float4*    best    = (const float4*)d_in[6];
    const float4*    best_gt = (const float4*)d_in[7];
    const long long* mapping = (const long long*)d_in[8];
    // d_in[9] = vector_dims == V (compile-time constant here)

    float* ws = (float*)d_ws;

    kld_kernel  <<<NB_KLD, BLK, 0, stream>>>(qy, ws + WS_KLD);
    recon_kernel<<<NB_REC, BLK, 0, stream>>>(zs, rzs, mapping, ws + WS_REC);
    pts_kernel  <<<NB_PTS, BLK, 0, stream>>>(pts, pts_gt, mapping,
                                             ws + WS_DISK, ws + WS_LAND);
    best_kernel <<<1, BLK, 0, stream>>>(best, best_gt, ws + WS_BEST);
    finalize_kernel<<<1, BLK, 0, stream>>>(ws, (float*)d_out);
}